// IouRegLoss_51221779972847
// MI455X (gfx1250) — compile-verified
//
#include <hip/hip_runtime.h>
#include <math.h>

typedef __attribute__((ext_vector_type(2))) float v2f;
typedef __attribute__((ext_vector_type(8))) float v8f;

// Sum of all 32 lane values using V_WMMA_F32_16X16X4_F32 with an all-ones B.
// A[m][0] = s_m (lanes 0-15), A[m][2] = s_{m+16} (lanes 16-31), other K zero.
// D[m][n] = s_m + s_{m+16} for every n. Lane<16 holds M=0..7 in d[0..7],
// lane>=16 holds M=8..15, so (sum of d[0..7]) + shfl_xor(.,16) = wave total
// in every lane. EXEC must be all ones at the call site.
__device__ __forceinline__ float wave_sum32_wmma(float x) {
    v2f a; a[0] = x;    a[1] = 0.0f;
    v2f b; b[0] = 1.0f; b[1] = 1.0f;   // all-ones B: layout-invariant
    v8f c = {0.0f, 0.0f, 0.0f, 0.0f, 0.0f, 0.0f, 0.0f, 0.0f};
    v8f d = __builtin_amdgcn_wmma_f32_16x16x4_f32(
        /*neg_a=*/false, a, /*neg_b=*/false, b,
        /*c_mod=*/(short)0, c, /*reuse_a=*/false, /*reuse_b=*/false);
    float t = ((d[0] + d[1]) + (d[2] + d[3])) + ((d[4] + d[5]) + (d[6] + d[7]));
    t += __shfl_xor(t, 16, 32);
    return t;
}

__global__ __launch_bounds__(256) void diou_partial_kernel(
    const float* __restrict__ pred, const float* __restrict__ gt,
    float* __restrict__ partial, int n)
{
    const int tid    = blockIdx.x * blockDim.x + threadIdx.x;
    const int stride = gridDim.x * blockDim.x;

    float acc = 0.0f;
    for (int i = tid; i < n; i += stride) {
        const float* p = pred + (size_t)i * 9;
        const float* g = gt   + (size_t)i * 9;

        // Streamed once, 288 MB > 192 MB L2: non-temporal loads.
        float px = __builtin_nontemporal_load(p + 0);
        float py = __builtin_nontemporal_load(p + 1);
        float pz = __builtin_nontemporal_load(p + 2);
        float pw = __builtin_nontemporal_load(p + 3);
        float pl = __builtin_nontemporal_load(p + 4);
        float ph = __builtin_nontemporal_load(p + 5);
        float pa = __builtin_nontemporal_load(p + 6);

        float gx = __builtin_nontemporal_load(g + 0);
        float gy = __builtin_nontemporal_load(g + 1);
        float gz = __builtin_nontemporal_load(g + 2);
        float gw = __builtin_nontemporal_load(g + 3);
        float gl = __builtin_nontemporal_load(g + 4);
        float gh = __builtin_nontemporal_load(g + 5);
        float ga = __builtin_nontemporal_load(g + 6);

        // Rotated corner0/corner2: center -/+ r, r = (hw*c + hl*s, hl*c - hw*s)
        float psn, pcs; sincosf(pa, &psn, &pcs);
        float gsn, gcs; sincosf(ga, &gsn, &gcs);

        float phw = 0.5f * pw, phl = 0.5f * pl;
        float ghw = 0.5f * gw, ghl = 0.5f * gl;

        float prx = phw * pcs + phl * psn;
        float pry = phl * pcs - phw * psn;
        float grx = ghw * gcs + ghl * gsn;
        float gry = ghl * gcs - ghw * gsn;

        float p0x = px - prx, p0y = py - pry;
        float p2x = px + prx, p2y = py + pry;
        float g0x = gx - grx, g0y = gy - gry;
        float g2x = gx + grx, g2y = gy + gry;

        // XY intersection / enclosing box
        float ix = fmaxf(fminf(p2x, g2x) - fmaxf(p0x, g0x), 0.0f);
        float iy = fmaxf(fminf(p2y, g2y) - fmaxf(p0y, g0y), 0.0f);
        float ox = fmaxf(fmaxf(p2x, g2x) - fminf(p0x, g0x), 0.0f);
        float oy = fmaxf(fmaxf(p2y, g2y) - fminf(p0y, g0y), 0.0f);

        // Z extent
        float ptop = pz + 0.5f * ph, pbot = pz - 0.5f * ph;
        float gtop = gz + 0.5f * gh, gbot = gz - 0.5f * gh;
        float ih = fmaxf(fminf(ptop, gtop) - fmaxf(pbot, gbot), 0.0f);
        float oh = fmaxf(fmaxf(ptop, gtop) - fminf(pbot, gbot), 0.0f);

        float ivol  = ix * iy * ih;
        float uvol  = pw * pl * ph + gw * gl * gh - ivol;

        float dx = gx - px, dy = gy - py, dz = gz - pz;
        float idiag = dx * dx + dy * dy + dz * dz;
        float odiag = ox * ox + oy * oy + oh * oh;

        float diou = ivol / uvol - idiag / odiag;
        diou = fminf(fmaxf(diou, -1.0f), 1.0f);
        acc += 1.0f - diou;
    }

    // Wave reduction on the matrix pipe (all 256 threads reconverged here).
    float wsum = wave_sum32_wmma(acc);

    __shared__ float lds_wsum[8];
    const int lane = threadIdx.x & 31;
    const int wid  = threadIdx.x >> 5;
    if (lane == 0) lds_wsum[wid] = wsum;
    __syncthreads();
    if (threadIdx.x == 0) {
        float s = 0.0f;
        #pragma unroll
        for (int k = 0; k < 8; ++k) s += lds_wsum[k];
        partial[blockIdx.x] = s;
    }
}

__global__ __launch_bounds__(32) void diou_finalize_kernel(
    const float* __restrict__ partial, int nb,
    float* __restrict__ out, float denom)
{
    // Single full wave: EXEC all ones; nb is a multiple of 32 so the loop
    // is divergence-free and lanes reconverge before the WMMA.
    float s = 0.0f;
    for (int i = (int)threadIdx.x; i < nb; i += 32)
        s += partial[i];
    float t = wave_sum32_wmma(s);
    if (threadIdx.x == 0) out[0] = t / denom;
}

extern "C" void kernel_launch(void* const* d_in, const int* in_sizes, int n_in,
                              void* d_out, int out_size, void* d_ws, size_t ws_size,
                              hipStream_t stream) {
    const float* pred = (const float*)d_in[0];
    const float* gt   = (const float*)d_in[1];
    const int n = in_sizes[0] / 9;

    const int NB = 1024;   // partials: 4 KB of workspace
    const int NT = 256;
    float* partial = (float*)d_ws;

    diou_partial_kernel<<<NB, NT, 0, stream>>>(pred, gt, partial, n);

    // Match JAX: float32 array / python float -> weak-typed f32 divisor
    const float denom = (float)((double)n + 0.0001);
    diou_finalize_kernel<<<1, 32, 0, stream>>>(partial, NB, (float*)d_out, denom);
}